// GroupGraph_68436008895084
// MI455X (gfx1250) — compile-verified
//
#include <hip/hip_runtime.h>
#include <math.h>

// ---------------------------------------------------------------------------
// GroupGraph (SGConv K=2 + sigmoid-gated attention pooling) for MI455X gfx1250
//   D = 512 (node dim), H = 64 (hidden). N,E,B,T derived from in_sizes.
// Big matmuls: V_WMMA_F32_16X16X4_F32 (fp32-faithful tensor path).
// LDS staging: TENSOR_LOAD_TO_LDS (TDM) with hardware padding + double buffer.
// ---------------------------------------------------------------------------

typedef float v2f __attribute__((ext_vector_type(2)));
typedef float v8f __attribute__((ext_vector_type(8)));
typedef unsigned int v4u __attribute__((ext_vector_type(4)));
typedef int v4i __attribute__((ext_vector_type(4)));
typedef int v8i __attribute__((ext_vector_type(8)));

#define DD 512
#define HH 64

// ---- TDM helper: 2-D tile (tile0 x tile1 elems of 4B) global -> LDS --------
// g1w0 packs {data_size=4B, pad_enable, pad_interval, pad_amount} per ISA 8.4.
__device__ __forceinline__ void tdm_load_2d(unsigned lds, unsigned long long ga,
                                            unsigned g1w0,
                                            unsigned tdim0, unsigned tdim1,
                                            unsigned tile0, unsigned tile1,
                                            unsigned stride) {
  v4u g0 = { 1u,                                   // count=1, user D#
             lds,                                  // lds_addr (bytes)
             (unsigned)ga,                         // global_addr[31:0]
             (unsigned)(ga >> 32) | (2u << 30) };  // global_addr[56:32] | type=2
  v8i g1 = { (int)g1w0,
             (int)((tdim0 & 0xffffu) << 16),                       // tensor_dim0 lo
             (int)(((tdim1 & 0xffffu) << 16) | (tdim0 >> 16)),     // td1 lo | td0 hi
             (int)(((tile0 & 0xffffu) << 16) | (tdim1 >> 16)),     // tile0 | td1 hi
             (int)(tile1 & 0xffffu),                               // tile1 (tile2=0)
             (int)stride,                                          // dim0_stride lo
             0, 0 };
  v4i z4 = { 0, 0, 0, 0 };
  v8i z8 = { 0, 0, 0, 0, 0, 0, 0, 0 };
  __builtin_amdgcn_tensor_load_to_lds(g0, g1, z4, z4, z8, 0);
}

// group1 word0 builders: data_size=4B (2<<16), optional LDS padding
#define G1W0_PAD16 ((2u << 16) | (1u << 20) | (3u << 22))  // +1 dword per 16 dwords
#define G1W0_PAD64 ((2u << 16) | (1u << 20) | (5u << 22))  // +1 dword per 64 dwords
#define G1W0_NOPAD (2u << 16)

// ---------------- small scan: node offsets + token starts -------------------
__global__ void scan_k(const int* __restrict__ node_num,
                       const int* __restrict__ seq_lens,
                       int* __restrict__ offsets, int* __restrict__ tokstart,
                       int B) {
  if (threadIdx.x == 0 && blockIdx.x == 0) {
    int no = 0, to = 0;
    for (int b = 0; b < B; ++b) {
      offsets[b]  = no; no += node_num[b];
      tokstart[b] = to; to += seq_lens[b];
    }
  }
}

// -------- per-token session id, global node index, last-token index ---------
__global__ void fill_k(const int* __restrict__ offsets,
                       const int* __restrict__ tokstart,
                       const int* __restrict__ seq_lens,
                       const int* __restrict__ sess_item_index,
                       int* __restrict__ tok2sess, int* __restrict__ glob_idx,
                       int* __restrict__ vn_idx) {
  int b = blockIdx.x;
  int s = tokstart[b], len = seq_lens[b], off = offsets[b];
  for (int i = threadIdx.x; i < len; i += blockDim.x) {
    int t = s + i;
    tok2sess[t] = b;
    glob_idx[t] = off + sess_item_index[t];
  }
  if (threadIdx.x == 0) vn_idx[b] = off + sess_item_index[s + len - 1];
}

// ---------------- degree (self-loop = 1) + edge increments ------------------
__global__ void deg_init_k(float* __restrict__ deg, int N) {
  int i = blockIdx.x * blockDim.x + threadIdx.x;
  if (i < N) deg[i] = 1.0f;
}
__global__ void deg_edge_k(const int* __restrict__ dst, float* __restrict__ deg, int E) {
  int e = blockIdx.x * blockDim.x + threadIdx.x;
  if (e < E) unsafeAtomicAdd(&deg[dst[e]], 1.0f);
}
__global__ void dinv_k(const float* __restrict__ deg, float* __restrict__ dinv, int N) {
  int i = blockIdx.x * blockDim.x + threadIdx.x;
  if (i < N) dinv[i] = rsqrtf(fmaxf(deg[i], 1.0f));
}

// -------- one propagation hop: y = D^-1/2 (A+I) D^-1/2 x --------------------
__global__ void hop_self_k(const float* __restrict__ x, const float* __restrict__ dinv,
                           float* __restrict__ y, int N) {
  long long idx = (long long)blockIdx.x * blockDim.x + threadIdx.x;
  long long tot = (long long)N * DD;
  if (idx < tot) {
    int node = (int)(idx >> 9);           // / 512
    float w = dinv[node]; w *= w;
    y[idx] = w * x[idx];
  }
}
__global__ void hop_edge_k(const int* __restrict__ src, const int* __restrict__ dst,
                           const float* __restrict__ dinv,
                           const float* __restrict__ x, float* __restrict__ y, int E) {
  int e    = blockIdx.x * (blockDim.x >> 5) + (threadIdx.x >> 5);
  int lane = threadIdx.x & 31;
  if (e >= E) return;
  int s = src[e], d = dst[e];
  float w = dinv[s] * dinv[d];
  const float* xs = x + (size_t)s * DD;
  float*       yd = y + (size_t)d * DD;
#pragma unroll
  for (int j = 0; j < DD / 32; ++j)
    unsafeAtomicAdd(&yd[lane + j * 32], w * xs[lane + j * 32]);
}

// -------- big GEMM: XW = Z[N,512] @ W_sg[512,512] + b_sg --------------------
// f32 WMMA; A/B tiles DMA'd by the Tensor Data Mover with hardware padding,
// double buffered so TDM of chunk k+1 overlaps WMMA of chunk k.
__global__ __launch_bounds__(256) void gemm_wsg_k(const float* __restrict__ Z,
                                                  const float* __restrict__ W,
                                                  const float* __restrict__ bias,
                                                  float* __restrict__ out, int N) {
  __shared__ float As[2][64][17];   // padded by TDM pad_interval=16dw,+1dw
  __shared__ float Bs[2][16][65];   // padded by TDM pad_interval=64dw,+1dw
  int tid  = threadIdx.x;
  int wave = tid >> 5, lane = tid & 31;
  int l15  = lane & 15, khalf = (lane >> 4) * 2;
  int bm = blockIdx.x * 64, bn = blockIdx.y * 64;
  int msub = (wave >> 1) * 16;        // 0,16,32,48
  int nsub = (wave & 1) * 32;         // 0,32  (each wave: 16x32 via 2 C tiles)
  v8f c0 = {}; v8f c1 = {};

  // prologue: DMA chunk 0 into buffer 0
  if (wave == 0) {
    tdm_load_2d((unsigned)(size_t)&As[0][0][0],
                (unsigned long long)(size_t)(Z + (size_t)bm * DD),
                G1W0_PAD16, DD, (unsigned)N, 16, 64, DD);
    tdm_load_2d((unsigned)(size_t)&Bs[0][0][0],
                (unsigned long long)(size_t)(W + bn),
                G1W0_PAD64, DD, DD, 64, 16, DD);
    __builtin_amdgcn_s_wait_tensorcnt(0);
  }
  __syncthreads();

  int cur = 0;
  for (int k0 = 0; k0 < DD; k0 += 16) {
    // kick off next chunk's DMA into the other buffer (overlaps WMMA below)
    if (wave == 0 && k0 + 16 < DD) {
      tdm_load_2d((unsigned)(size_t)&As[cur ^ 1][0][0],
                  (unsigned long long)(size_t)(Z + (size_t)bm * DD + k0 + 16),
                  G1W0_PAD16, DD, (unsigned)N, 16, 64, DD);
      tdm_load_2d((unsigned)(size_t)&Bs[cur ^ 1][0][0],
                  (unsigned long long)(size_t)(W + (size_t)(k0 + 16) * DD + bn),
                  G1W0_PAD64, DD, DD, 64, 16, DD);
    }
#pragma unroll
    for (int ks = 0; ks < 4; ++ks) {
      int kl = ks * 4 + khalf;
      v2f a;  a.x  = As[cur][msub + l15][kl];      a.y  = As[cur][msub + l15][kl + 1];
      v2f b0; b0.x = Bs[cur][kl][nsub + l15];      b0.y = Bs[cur][kl + 1][nsub + l15];
      v2f b1; b1.x = Bs[cur][kl][nsub + 16 + l15]; b1.y = Bs[cur][kl + 1][nsub + 16 + l15];
      c0 = __builtin_amdgcn_wmma_f32_16x16x4_f32(false, a, false, b0, (short)0, c0, false, false);
      c1 = __builtin_amdgcn_wmma_f32_16x16x4_f32(false, a, false, b1, (short)0, c1, false, false);
    }
    if (wave == 0) __builtin_amdgcn_s_wait_tensorcnt(0);
    __syncthreads();
    cur ^= 1;
  }

  int col0 = bn + nsub + l15, col1 = col0 + 16;
  float bi0 = bias[col0], bi1 = bias[col1];
  int rhi = (lane >> 4) * 8;
#pragma unroll
  for (int r = 0; r < 8; ++r) {
    int row = bm + msub + rhi + r;
    out[(size_t)row * DD + col0] = c0[r] + bi0;
    out[(size_t)row * DD + col1] = c1[r] + bi1;
  }
}

// -------- Pre[b,h] = b1[h]+b2[h] + v_n[b,:] @ W1[:,h]  (tiny) ---------------
__global__ __launch_bounds__(64) void pre_k(const float* __restrict__ XW,
                                            const int* __restrict__ vn_idx,
                                            const float* __restrict__ W1,
                                            const float* __restrict__ b1,
                                            const float* __restrict__ b2,
                                            float* __restrict__ Pre) {
  __shared__ float row[DD];
  int b = blockIdx.x, h = threadIdx.x;
  const float* vrow = XW + (size_t)vn_idx[b] * DD;
  for (int i = h; i < DD; i += 64) row[i] = vrow[i];
  __syncthreads();
  float acc = b1[h] + b2[h];
  for (int d = 0; d < DD; ++d) acc += row[d] * W1[d * HH + h];
  Pre[b * HH + h] = acc;
}

// -------- gate+alpha fused: alpha[t] = qb + sum_h qw[h]*sigmoid(tok@W2 + Pre)
// Whole W2 (512x64 fp32 = 128 KB) DMA'd once into the 320 KB LDS by the TDM.
__global__ __launch_bounds__(256) void gate_alpha_k(const float* __restrict__ XW,
                                                    const float* __restrict__ W2,
                                                    const float* __restrict__ qw,
                                                    const float* __restrict__ qb,
                                                    const int* __restrict__ glob_idx,
                                                    const int* __restrict__ tok2sess,
                                                    const float* __restrict__ Pre,
                                                    float* __restrict__ alpha, int T) {
  extern __shared__ float Ws[];                    // [512][64]
  int tid = threadIdx.x;
  if (tid < 32) {
    // one 1-D TDM: 32768 contiguous dwords global -> LDS
    tdm_load_2d((unsigned)(size_t)&Ws[0], (unsigned long long)(size_t)W2,
                G1W0_NOPAD, (unsigned)(DD * HH), 1, (unsigned)(DD * HH), 1,
                (unsigned)(DD * HH));
    __builtin_amdgcn_s_wait_tensorcnt(0);
  }
  __syncthreads();
  int lane = tid & 31, wave = tid >> 5;
  int l15 = lane & 15, khalf = (lane >> 4) * 2;
  float qbv = qb[0];
  float qw0 = qw[l15], qw1 = qw[16 + l15], qw2 = qw[32 + l15], qw3 = qw[48 + l15];
  int nTiles = T >> 4;
  int wgid = blockIdx.x * 8 + wave, nW = gridDim.x * 8;
  for (int tile = wgid; tile < nTiles; tile += nW) {
    int t0 = tile * 16;
    const float* rowp = XW + (size_t)glob_idx[t0 + l15] * DD;
    v8f c0 = {}, c1 = {}, c2 = {}, c3 = {};
    for (int k0 = 0; k0 < DD; k0 += 4) {
      int k = k0 + khalf;
      v2f a = *(const v2f*)(rowp + k);             // gathered, 8B aligned
      v2f b0, b1, b2, b3;
      b0.x = Ws[k * HH + l15];       b0.y = Ws[(k + 1) * HH + l15];
      b1.x = Ws[k * HH + 16 + l15];  b1.y = Ws[(k + 1) * HH + 16 + l15];
      b2.x = Ws[k * HH + 32 + l15];  b2.y = Ws[(k + 1) * HH + 32 + l15];
      b3.x = Ws[k * HH + 48 + l15];  b3.y = Ws[(k + 1) * HH + 48 + l15];
      c0 = __builtin_amdgcn_wmma_f32_16x16x4_f32(false, a, false, b0, (short)0, c0, false, false);
      c1 = __builtin_amdgcn_wmma_f32_16x16x4_f32(false, a, false, b1, (short)0, c1, false, false);
      c2 = __builtin_amdgcn_wmma_f32_16x16x4_f32(false, a, false, b2, (short)0, c2, false, false);
      c3 = __builtin_amdgcn_wmma_f32_16x16x4_f32(false, a, false, b3, (short)0, c3, false, false);
    }
    int rhi = (lane >> 4) * 8;
#pragma unroll
    for (int r = 0; r < 8; ++r) {
      int tok = t0 + rhi + r;
      const float* pre = Pre + (size_t)tok2sess[tok] * HH;
      float v = qw0 / (1.0f + __expf(-(c0[r] + pre[l15])))
              + qw1 / (1.0f + __expf(-(c1[r] + pre[16 + l15])))
              + qw2 / (1.0f + __expf(-(c2[r] + pre[32 + l15])))
              + qw3 / (1.0f + __expf(-(c3[r] + pre[48 + l15])));
      // reduce the 16 h-lanes of each half-wave (wave32)
      v += __shfl_xor(v, 1, 32);
      v += __shfl_xor(v, 2, 32);
      v += __shfl_xor(v, 4, 32);
      v += __shfl_xor(v, 8, 32);
      if (l15 == 0) alpha[tok] = qbv + v;
    }
  }
}

// -------- s_g[b,:] = sum_t alpha[t] * tokens[t,:] ---------------------------
__global__ __launch_bounds__(256) void sg_k(const float* __restrict__ XW,
                                            const float* __restrict__ alpha,
                                            const int* __restrict__ glob_idx,
                                            const int* __restrict__ tokstart,
                                            const int* __restrict__ seq_lens,
                                            float* __restrict__ sg) {
  int b = blockIdx.x, tid = threadIdx.x;
  int s = tokstart[b], e = s + seq_lens[b];
  float a0 = 0.f, a1 = 0.f;
  for (int t = s; t < e; ++t) {
    float al = alpha[t];
    const float* row = XW + (size_t)glob_idx[t] * DD;
    a0 += al * row[tid];
    a1 += al * row[tid + 256];
  }
  sg[(size_t)b * DD + tid]       = a0;
  sg[(size_t)b * DD + tid + 256] = a1;
}

// -------- h_s = [v_n, s_g] @ W3 + b3 ----------------------------------------
__global__ __launch_bounds__(64) void final_k(const float* __restrict__ XW,
                                              const int* __restrict__ vn_idx,
                                              const float* __restrict__ sg,
                                              const float* __restrict__ W3,
                                              const float* __restrict__ b3,
                                              float* __restrict__ out) {
  __shared__ float vn[DD], sr[DD];
  int b = blockIdx.x, h = threadIdx.x;
  const float* vrow = XW + (size_t)vn_idx[b] * DD;
  const float* srow = sg + (size_t)b * DD;
  for (int i = h; i < DD; i += 64) { vn[i] = vrow[i]; sr[i] = srow[i]; }
  __syncthreads();
  float acc = b3[h];
  for (int d = 0; d < DD; ++d) acc += vn[d] * W3[d * HH + h];
  for (int d = 0; d < DD; ++d) acc += sr[d] * W3[(DD + d) * HH + h];
  out[(size_t)b * HH + h] = acc;
}

// ---------------------------------------------------------------------------
extern "C" void kernel_launch(void* const* d_in, const int* in_sizes, int n_in,
                              void* d_out, int out_size, void* d_ws, size_t ws_size,
                              hipStream_t stream) {
  const float* hidden = (const float*)d_in[0];
  const int*   eidx   = (const int*)d_in[1];
  const int*   nodenum= (const int*)d_in[2];
  const int*   seqlen = (const int*)d_in[3];
  const int*   sidx   = (const int*)d_in[4];
  const float* W_sg   = (const float*)d_in[5];
  const float* b_sg   = (const float*)d_in[6];
  const float* W1     = (const float*)d_in[7];
  const float* b1     = (const float*)d_in[8];
  const float* W2     = (const float*)d_in[9];
  const float* b2     = (const float*)d_in[10];
  const float* qw     = (const float*)d_in[11];
  const float* qb     = (const float*)d_in[12];
  const float* W3     = (const float*)d_in[13];
  const float* b3     = (const float*)d_in[14];
  float* out = (float*)d_out;

  const int N = in_sizes[0] / DD;
  const int E = in_sizes[1] / 2;
  const int B = in_sizes[2];
  const int T = in_sizes[4];
  const int* esrc = eidx;          // edge_index[0]
  const int* edst = eidx + E;      // edge_index[1]

  // workspace layout
  char* p = (char*)d_ws;
  float* deg  = (float*)p;                 p += (size_t)N * 4;
  float* dinv = (float*)p;                 p += (size_t)N * 4;
  float* y1   = (float*)p;                 p += (size_t)N * DD * 4;
  float* y2   = (float*)p;                 p += (size_t)N * DD * 4;
  float* xw   = y1;                        // reuse: y1 dead once y2 exists
  float* Pre  = (float*)p;                 p += (size_t)B * HH * 4;
  float* alpha= (float*)p;                 p += (size_t)T * 4;
  float* sg   = (float*)p;                 p += (size_t)B * DD * 4;
  int* offsets = (int*)p;                  p += (size_t)B * 4;
  int* tokstart= (int*)p;                  p += (size_t)B * 4;
  int* tok2sess= (int*)p;                  p += (size_t)T * 4;
  int* glob    = (int*)p;                  p += (size_t)T * 4;
  int* vn_idx  = (int*)p;                  p += (size_t)B * 4;

  // 1) scans + token maps
  scan_k<<<1, 32, 0, stream>>>(nodenum, seqlen, offsets, tokstart, B);
  fill_k<<<B, 128, 0, stream>>>(offsets, tokstart, seqlen, sidx, tok2sess, glob, vn_idx);

  // 2) degrees + symmetric norm
  deg_init_k<<<(N + 255) / 256, 256, 0, stream>>>(deg, N);
  deg_edge_k<<<(E + 255) / 256, 256, 0, stream>>>(edst, deg, E);
  dinv_k<<<(N + 255) / 256, 256, 0, stream>>>(deg, dinv, N);

  // 3) two propagation hops (self-loop init + edge scatter-add; L2 resident)
  long long tot = (long long)N * DD;
  int selfBlocks = (int)((tot + 255) / 256);
  int edgeBlocks = (E + 7) / 8;                       // 8 waves/block, wave/edge
  hop_self_k<<<selfBlocks, 256, 0, stream>>>(hidden, dinv, y1, N);
  hop_edge_k<<<edgeBlocks, 256, 0, stream>>>(esrc, edst, dinv, hidden, y1, E);
  hop_self_k<<<selfBlocks, 256, 0, stream>>>(y1, dinv, y2, N);
  hop_edge_k<<<edgeBlocks, 256, 0, stream>>>(esrc, edst, dinv, y1, y2, E);

  // 4) XW = y2 @ W_sg + b_sg   (f32 WMMA, TDM-staged 64x64 block tiles)
  dim3 ggrid(N / 64, DD / 64);
  gemm_wsg_k<<<ggrid, 256, 0, stream>>>(y2, W_sg, b_sg, xw, N);

  // 5) Pre = v_n @ W1 + b1 + b2
  pre_k<<<B, 64, 0, stream>>>(xw, vn_idx, W1, b1, b2, Pre);

  // 6) fused gate matmul + sigmoid + qw-dot -> alpha (f32 WMMA, W2 via TDM)
  gate_alpha_k<<<64, 256, DD * HH * sizeof(float), stream>>>(
      xw, W2, qw, qb, glob, tok2sess, Pre, alpha, T);

  // 7) attention pooling per session
  sg_k<<<B, 256, 0, stream>>>(xw, alpha, glob, tokstart, seqlen, sg);

  // 8) h_s = [v_n, s_g] @ W3 + b3
  final_k<<<B, 64, 0, stream>>>(xw, vn_idx, sg, W3, b3, out);
}